// CorrectSemanticGNN_26053271617836
// MI455X (gfx1250) — compile-verified
//
#include <hip/hip_runtime.h>
#include <math.h>

#define DIN 135
#define TD 9
#define NHEADS 4
#define ENC_NEG_INF 0x007FFFFFu

typedef float v2f __attribute__((ext_vector_type(2)));
typedef float v8f __attribute__((ext_vector_type(8)));

__device__ __forceinline__ unsigned encf(float v) {
  unsigned b = __float_as_uint(v);
  return (b & 0x80000000u) ? ~b : (b | 0x80000000u);
}
__device__ __forceinline__ float decf(unsigned s) {
  unsigned b = (s & 0x80000000u) ? (s & 0x7FFFFFFFu) : ~s;
  return __uint_as_float(b);
}
__device__ __forceinline__ float leaky(float v) { return v < 0.f ? 0.2f * v : v; }

// ---------------- fill kernels (graph-capture-safe init) ----------------
__global__ void k_fill_f32(float* p, float v, size_t n) {
  size_t i = (size_t)blockIdx.x * blockDim.x + threadIdx.x;
  if (i < n) p[i] = v;
}
__global__ void k_fill_u32(unsigned* p, unsigned v, size_t n) {
  size_t i = (size_t)blockIdx.x * blockDim.x + threadIdx.x;
  if (i < n) p[i] = v;
}

// ---------------- edge-similarity scale ----------------
__global__ void k_scale_sum(const float* __restrict__ x, const int* __restrict__ ei,
                            int E, float* accum) {
  int e = blockIdx.x * blockDim.x + threadIdx.x;
  float sim = 0.f;
  if (e < E) {
    int s = ei[e], d = ei[E + e];
    const float* ti = x + (size_t)s * DIN + (DIN - TD);
    const float* tj = x + (size_t)d * DIN + (DIN - TD);
    float dot = 0.f, ni = 0.f, nj = 0.f;
#pragma unroll
    for (int k = 0; k < TD; ++k) {
      float a = ti[k], b = tj[k];
      dot += a * b; ni += a * a; nj += b * b;
    }
    sim = dot / (fmaxf(sqrtf(ni), 1e-8f) * fmaxf(sqrtf(nj), 1e-8f));
  }
  for (int off = 16; off; off >>= 1) sim += __shfl_xor(sim, off, 32);
  if ((threadIdx.x & 31) == 0) atomicAdd(accum, sim);
}
__global__ void k_scale_fin(const float* accum, const float* tw, float invE, float* scale) {
  scale[0] = 1.1f + 0.1f * tw[0] * (accum[0] * invE);
}

// ---------------- fp32 WMMA GEMM:  C[M x Nn] = A[M x K] @ B[K x Nn] (+bias) ----------------
// One wave computes a 16x64 strip (4 N-tiles) using V_WMMA_F32_16X16X4_F32.
// KMOD4=true  : K % 4 == 0, rows 8B-aligned -> unpredicated loads, b64 A-fragment load.
// KMOD4=false : arbitrary K (input projection, K=135) -> predicated tail loads.
template <bool KMOD4>
__global__ __launch_bounds__(256) void k_gemm_wmma(
    const float* __restrict__ A, const float* __restrict__ B,
    const float* __restrict__ bias, float* __restrict__ C,
    int M, int K, int Nn) {
  int lane = threadIdx.x & 31;
  int wid = threadIdx.x >> 5;
  int strips_n = Nn >> 6;
  int strip = blockIdx.x * 8 + wid;
  int total = (M >> 4) * strips_n;
  if (strip >= total) return;
  int tm = strip / strips_n;
  int tn = strip % strips_n;
  int m0 = tm << 4;
  int n0 = tn << 6;
  int lrow = lane & 15;
  int lhalf = lane >> 4;

  v8f acc0 = {}; v8f acc1 = {}; v8f acc2 = {}; v8f acc3 = {};
  const float* Arow = A + (size_t)(m0 + lrow) * K;

  if (KMOD4) {
    const v2f* Arow2 = (const v2f*)Arow;
    for (int k = 0; k < K; k += 4) {
      if (k + 32 < K) __builtin_prefetch(Arow + k + 32, 0, 3);
      v2f av = Arow2[(k >> 1) + lhalf];                 // global_load_b64
      const float* B0 = B + (size_t)(k + (lhalf << 1)) * Nn + n0 + lrow;
      const float* B1 = B0 + Nn;
      float b00 = B0[0], b01 = B0[16], b02 = B0[32], b03 = B0[48];
      float b10 = B1[0], b11 = B1[16], b12 = B1[32], b13 = B1[48];
      v2f bv0 = {b00, b10}; v2f bv1 = {b01, b11}; v2f bv2 = {b02, b12}; v2f bv3 = {b03, b13};
      acc0 = __builtin_amdgcn_wmma_f32_16x16x4_f32(false, av, false, bv0, (short)0, acc0, false, false);
      acc1 = __builtin_amdgcn_wmma_f32_16x16x4_f32(false, av, false, bv1, (short)0, acc1, false, false);
      acc2 = __builtin_amdgcn_wmma_f32_16x16x4_f32(false, av, false, bv2, (short)0, acc2, false, false);
      acc3 = __builtin_amdgcn_wmma_f32_16x16x4_f32(false, av, false, bv3, (short)0, acc3, false, false);
    }
  } else {
    for (int k = 0; k < K; k += 4) {
      int ka = k + (lhalf << 1);
      if (k + 32 < K) __builtin_prefetch(Arow + k + 32, 0, 3);
      float a0 = 0.f, a1 = 0.f;
      if (ka < K)     a0 = Arow[ka];
      if (ka + 1 < K) a1 = Arow[ka + 1];
      v2f av = {a0, a1};
      float b00 = 0.f, b01 = 0.f, b02 = 0.f, b03 = 0.f;
      float b10 = 0.f, b11 = 0.f, b12 = 0.f, b13 = 0.f;
      const float* B0 = B + (size_t)ka * Nn + n0 + lrow;
      if (ka < K)     { b00 = B0[0];  b01 = B0[16];  b02 = B0[32];  b03 = B0[48]; }
      if (ka + 1 < K) { const float* B1 = B0 + Nn;
                        b10 = B1[0];  b11 = B1[16];  b12 = B1[32];  b13 = B1[48]; }
      v2f bv0 = {b00, b10}; v2f bv1 = {b01, b11}; v2f bv2 = {b02, b12}; v2f bv3 = {b03, b13};
      acc0 = __builtin_amdgcn_wmma_f32_16x16x4_f32(false, av, false, bv0, (short)0, acc0, false, false);
      acc1 = __builtin_amdgcn_wmma_f32_16x16x4_f32(false, av, false, bv1, (short)0, acc1, false, false);
      acc2 = __builtin_amdgcn_wmma_f32_16x16x4_f32(false, av, false, bv2, (short)0, acc2, false, false);
      acc3 = __builtin_amdgcn_wmma_f32_16x16x4_f32(false, av, false, bv3, (short)0, acc3, false, false);
    }
  }

  int col0 = n0 + lrow;
  int rbase = m0 + (lhalf ? 8 : 0);
  float bs0 = bias ? bias[col0]      : 0.f;
  float bs1 = bias ? bias[col0 + 16] : 0.f;
  float bs2 = bias ? bias[col0 + 32] : 0.f;
  float bs3 = bias ? bias[col0 + 48] : 0.f;
#pragma unroll
  for (int r = 0; r < 8; ++r) {
    float* Cr = C + (size_t)(rbase + r) * Nn;
    Cr[col0]      = acc0[r] + bs0;
    Cr[col0 + 16] = acc1[r] + bs1;
    Cr[col0 + 32] = acc2[r] + bs2;
    Cr[col0 + 48] = acc3[r] + bs3;
  }
}

// ---------------- attention source/dest dots: one wave per (node,head) ----------------
__global__ __launch_bounds__(256) void k_dots(
    const float* __restrict__ hh, const float* __restrict__ as, const float* __restrict__ ad,
    float* __restrict__ es, float* __restrict__ ed, int n, int C) {
  int lane = threadIdx.x & 31;
  int w = blockIdx.x * 8 + (threadIdx.x >> 5);
  if (w >= n * NHEADS) return;
  int node = w / NHEADS, head = w % NHEADS;
  const float* hp = hh + (size_t)node * NHEADS * C + (size_t)head * C;
  const float* ap = as + head * C;
  const float* dp = ad + head * C;
  float s = 0.f, d = 0.f;
  for (int i = lane; i < C; i += 32) {
    float v = hp[i];
    s += v * ap[i];
    d += v * dp[i];
  }
  for (int off = 16; off; off >>= 1) {
    s += __shfl_xor(s, off, 32);
    d += __shfl_xor(d, off, 32);
  }
  if (lane == 0) { es[w] = s; ed[w] = d; }
}

// ---------------- softmax pass 1: per-dst max (order-preserving uint atomicMax) ----------------
__global__ void k_edge_max(const float* __restrict__ es, const float* __restrict__ ed,
                           const int* __restrict__ ei, unsigned* __restrict__ menc,
                           int E, int EL) {
  int e = blockIdx.x * blockDim.x + threadIdx.x;
  if (e >= EL) return;
  int s, d;
  if (e < E) { s = ei[e]; d = ei[E + e]; } else { s = d = e - E; }
#pragma unroll
  for (int h = 0; h < NHEADS; ++h) {
    float v = leaky(es[(size_t)s * NHEADS + h] + ed[(size_t)d * NHEADS + h]);
    atomicMax(menc + (size_t)d * NHEADS + h, encf(v));
  }
}

// ---------------- softmax pass 2: exp(e - m), accumulate denominator ----------------
__global__ void k_edge_exp(const float* __restrict__ es, const float* __restrict__ ed,
                           const int* __restrict__ ei, const unsigned* __restrict__ menc,
                           float* __restrict__ den, float* __restrict__ ex,
                           int E, int EL) {
  int e = blockIdx.x * blockDim.x + threadIdx.x;
  if (e >= EL) return;
  int s, d;
  if (e < E) { s = ei[e]; d = ei[E + e]; } else { s = d = e - E; }
#pragma unroll
  for (int h = 0; h < NHEADS; ++h) {
    float v = leaky(es[(size_t)s * NHEADS + h] + ed[(size_t)d * NHEADS + h]);
    float m = decf(menc[(size_t)d * NHEADS + h]);
    float xv = expf(v - m);
    ex[(size_t)e * NHEADS + h] = xv;
    atomicAdd(den + (size_t)d * NHEADS + h, xv);
  }
}

// ---------------- softmax pass 3: scatter h[src] * alpha into dst accumulator ----------------
__global__ __launch_bounds__(256) void k_edge_scatter(
    const float* __restrict__ hh, const int* __restrict__ ei,
    const float* __restrict__ ex, const float* __restrict__ den,
    float* __restrict__ outp, int E, int EL, int C) {
  int lane = threadIdx.x & 31;
  int e = blockIdx.x * 8 + (threadIdx.x >> 5);
  if (e >= EL) return;
  int s, d;
  if (e < E) { s = ei[e]; d = ei[E + e]; } else { s = d = e - E; }
#pragma unroll
  for (int h = 0; h < NHEADS; ++h) {
    float alpha = ex[(size_t)e * NHEADS + h] / (den[(size_t)d * NHEADS + h] + 1e-16f);
    const float* hs = hh + (size_t)s * NHEADS * C + (size_t)h * C;
    float* od = outp + (size_t)d * NHEADS * C + (size_t)h * C;
    for (int i = lane; i < C; i += 32) atomicAdd(od + i, hs[i] * alpha);
  }
}

// ---------------- fused mean-heads + bias (+scale) + ReLU + residual + LayerNorm ----------------
// one wave per node; C in {64,128}
__global__ __launch_bounds__(256) void k_combine(
    const float* __restrict__ attn, const float* __restrict__ res,
    const float* __restrict__ bvec, const float* __restrict__ scale_ptr,
    const float* __restrict__ g, const float* __restrict__ bln,
    float* __restrict__ hout, int n, int C) {
  int lane = threadIdx.x & 31;
  int node = blockIdx.x * 8 + (threadIdx.x >> 5);
  if (node >= n) return;
  float sc = scale_ptr ? scale_ptr[0] : 1.0f;
  int cpl = C >> 5;
  float v[4];
  float s = 0.f, s2 = 0.f;
  const float* an = attn + (size_t)node * NHEADS * C;
  const float* rn = res + (size_t)node * C;
  for (int j = 0; j < cpl; ++j) {
    int f = lane + 32 * j;
    float a = 0.f;
#pragma unroll
    for (int h = 0; h < NHEADS; ++h) a += an[(size_t)h * C + f];
    a = (a * (1.0f / NHEADS) + bvec[f]) * sc;
    float xv = fmaxf(a, 0.f) + rn[f];
    v[j] = xv; s += xv; s2 += xv * xv;
  }
  for (int off = 16; off; off >>= 1) {
    s += __shfl_xor(s, off, 32);
    s2 += __shfl_xor(s2, off, 32);
  }
  float mu = s / (float)C;
  float var = s2 / (float)C - mu * mu;
  float inv = 1.0f / sqrtf(var + 1e-5f);
  float* hn = hout + (size_t)node * C;
  for (int j = 0; j < cpl; ++j) {
    int f = lane + 32 * j;
    hn[f] = (v[j] - mu) * inv * g[f] + bln[f];
  }
}

// ---------------- final classifier on node 0 ----------------
__global__ void k_final(const float* __restrict__ h, const float* __restrict__ c1W,
                        const float* __restrict__ c1b, const float* __restrict__ clg,
                        const float* __restrict__ clb, const float* __restrict__ c2W,
                        const float* __restrict__ c2b, float* __restrict__ out) {
  __shared__ float ego[64];
  __shared__ float z[64];
  __shared__ float zn[64];
  __shared__ float red[2];
  int t = threadIdx.x;
  ego[t] = h[t];
  __syncthreads();
  float acc = c1b[t];
  for (int k = 0; k < 64; ++k) acc += ego[k] * c1W[k * 64 + t];
  z[t] = fmaxf(acc, 0.f);
  __syncthreads();
  if (t == 0) {
    float s = 0.f, s2 = 0.f;
    for (int j = 0; j < 64; ++j) { s += z[j]; s2 += z[j] * z[j]; }
    float mu = s / 64.f;
    float var = s2 / 64.f - mu * mu;
    red[0] = mu;
    red[1] = 1.0f / sqrtf(var + 1e-5f);
  }
  __syncthreads();
  zn[t] = (z[t] - red[0]) * red[1] * clg[t] + clb[t];
  __syncthreads();
  if (t < 2) {
    float o = c2b[t];
    for (int j = 0; j < 64; ++j) o += zn[j] * c2W[j * 2 + t];
    out[t] = o;
  }
}

// ---------------- host orchestration ----------------
extern "C" void kernel_launch(void* const* d_in, const int* in_sizes, int n_in,
                              void* d_out, int out_size, void* d_ws, size_t ws_size,
                              hipStream_t stream) {
  (void)n_in; (void)out_size; (void)ws_size;
  const float* x    = (const float*)d_in[0];
  const int*   ei   = (const int*)d_in[1];
  const float* W_in = (const float*)d_in[3];
  const float* b_in = (const float*)d_in[4];
  const float* g1W  = (const float*)d_in[5];
  const float* g1as = (const float*)d_in[6];
  const float* g1ad = (const float*)d_in[7];
  const float* g1b  = (const float*)d_in[8];
  const float* g2W  = (const float*)d_in[9];
  const float* g2as = (const float*)d_in[10];
  const float* g2ad = (const float*)d_in[11];
  const float* g2b  = (const float*)d_in[12];
  const float* g3W  = (const float*)d_in[13];
  const float* g3as = (const float*)d_in[14];
  const float* g3ad = (const float*)d_in[15];
  const float* g3b  = (const float*)d_in[16];
  const float* r1W  = (const float*)d_in[17];
  const float* r1b  = (const float*)d_in[18];
  const float* r2W  = (const float*)d_in[19];
  const float* r2b  = (const float*)d_in[20];
  const float* r3W  = (const float*)d_in[21];
  const float* r3b  = (const float*)d_in[22];
  const float* n1g  = (const float*)d_in[23];
  const float* n1b  = (const float*)d_in[24];
  const float* n2g  = (const float*)d_in[25];
  const float* n2b  = (const float*)d_in[26];
  const float* n3g  = (const float*)d_in[27];
  const float* n3b  = (const float*)d_in[28];
  const float* tw   = (const float*)d_in[29];
  const float* c1W  = (const float*)d_in[30];
  const float* c1b  = (const float*)d_in[31];
  const float* clg  = (const float*)d_in[32];
  const float* clb  = (const float*)d_in[33];
  const float* c2W  = (const float*)d_in[34];
  const float* c2b  = (const float*)d_in[35];

  const int N  = in_sizes[0] / DIN;   // 50000
  const int E  = in_sizes[1] / 2;     // 500000
  const int EL = E + N;               // edges + self loops

  // workspace layout
  char* base = (char*)d_ws;
  size_t off = 0;
  auto take = [&](size_t elems) -> char* {
    char* p = base + off;
    off += ((elems * 4 + 255) / 256) * 256;
    return p;
  };
  float*    hA   = (float*)take((size_t)N * 128);
  float*    hB   = (float*)take((size_t)N * 128);
  float*    resb = (float*)take((size_t)N * 128);
  float*    hh   = (float*)take((size_t)N * 512);
  float*    attn = (float*)take((size_t)N * 512);
  float*    es   = (float*)take((size_t)N * NHEADS);
  float*    edv  = (float*)take((size_t)N * NHEADS);
  unsigned* menc = (unsigned*)take((size_t)N * NHEADS);
  float*    den  = (float*)take((size_t)N * NHEADS);
  float*    exb  = (float*)take((size_t)EL * NHEADS);
  float*    misc = (float*)take(8);   // [0]=accum, [1]=scale

  // ---- similarity scale ----
  k_fill_f32<<<1, 32, 0, stream>>>(misc, 0.f, 2);
  k_scale_sum<<<(E + 255) / 256, 256, 0, stream>>>(x, ei, E, misc);
  k_scale_fin<<<1, 1, 0, stream>>>(misc, tw, 1.0f / (float)E, misc + 1);

  // ---- input projection: hA = x @ W_in + b_in  (K=135 -> predicated path) ----
  {
    int strips = (N / 16) * (128 / 64);
    k_gemm_wmma<false><<<(strips + 7) / 8, 256, 0, stream>>>(x, W_in, b_in, hA, N, DIN, 128);
  }

  struct LayerP {
    const float *gW, *gas, *gad, *gb, *rW, *rb, *ng, *nb;
    int C, use_scale;
  };
  LayerP L[3] = {
    { g1W, g1as, g1ad, g1b, r1W, r1b, n1g, n1b, 128, 1 },
    { g2W, g2as, g2ad, g2b, r2W, r2b, n2g, n2b, 128, 1 },
    { g3W, g3as, g3ad, g3b, r3W, r3b, n3g, n3b, 64, 0 },
  };

  float* cur = hA;
  float* nxt = hB;
  for (int li = 0; li < 3; ++li) {
    int C = L[li].C;
    int HC = NHEADS * C;

    // head features: hh = cur @ gW   (K=128 -> fast path)
    {
      int strips = (N / 16) * (HC / 64);
      k_gemm_wmma<true><<<(strips + 7) / 8, 256, 0, stream>>>(cur, L[li].gW, nullptr, hh, N, 128, HC);
    }
    // attention logits per (node, head)
    k_dots<<<(N * NHEADS + 7) / 8, 256, 0, stream>>>(hh, L[li].gas, L[li].gad, es, edv, N, C);

    // init per-dst max / denom / output accumulator
    k_fill_u32<<<((size_t)N * NHEADS + 255) / 256, 256, 0, stream>>>(menc, ENC_NEG_INF, (size_t)N * NHEADS);
    k_fill_f32<<<((size_t)N * NHEADS + 255) / 256, 256, 0, stream>>>(den, 0.f, (size_t)N * NHEADS);
    k_fill_f32<<<((size_t)N * HC + 255) / 256, 256, 0, stream>>>(attn, 0.f, (size_t)N * HC);

    k_edge_max<<<(EL + 255) / 256, 256, 0, stream>>>(es, edv, ei, menc, E, EL);
    k_edge_exp<<<(EL + 255) / 256, 256, 0, stream>>>(es, edv, ei, menc, den, exb, E, EL);
    k_edge_scatter<<<(EL + 7) / 8, 256, 0, stream>>>(hh, ei, exb, den, attn, E, EL, C);

    // residual path: resb = cur @ rW + rb   (K=128 -> fast path)
    {
      int strips = (N / 16) * (C / 64);
      k_gemm_wmma<true><<<(strips + 7) / 8, 256, 0, stream>>>(cur, L[li].rW, L[li].rb, resb, N, 128, C);
    }
    // fused epilogue -> nxt
    k_combine<<<(N + 7) / 8, 256, 0, stream>>>(attn, resb, L[li].gb,
                                               L[li].use_scale ? (misc + 1) : nullptr,
                                               L[li].ng, L[li].nb, nxt, N, C);
    float* t = cur; cur = nxt; nxt = t;
  }

  // classifier head on node 0
  k_final<<<1, 64, 0, stream>>>(cur, c1W, c1b, clg, clb, c2W, c2b, (float*)d_out);
}